// TGCN_85538568667147
// MI455X (gfx1250) — compile-verified
//
#include <hip/hip_runtime.h>

// ---------------------------------------------------------------------------
// TGCN forward for MI455X (gfx1250), fp32 via V_WMMA_F32_16X16X4_F32 with
// ASYNCcnt-tracked GLOBAL_LOAD_ASYNC_TO_LDS double-buffered staging.
//
//   out = ReLU( X W + S0 Ws0 + mode1( X Ws1 + S0 Ws01 ) + (b+bs0+bs1+bs01) )
//   S0[b,k,j,d] = sum_i adj0[i,k] X[b,i,j,d]         (kernel 1, 68.7 GFLOP)
//   mode1(Q)[b,i,l,e] = sum_j adj1[j,l] Q[b,i,j,e]   (kernel 2, fused)
// ---------------------------------------------------------------------------

typedef float v2f __attribute__((ext_vector_type(2)));
typedef float v8f __attribute__((ext_vector_type(8)));
typedef int   v4i __attribute__((vector_size(16)));   // matches builtin param type

#define WMMA_F32(a, b, c) \
  __builtin_amdgcn_wmma_f32_16x16x4_f32(false, (a), false, (b), (short)0, (c), false, false)

// Async global->LDS copies (tracked by ASYNCcnt, no VGPR round-trip).
#define GASI(p) ((__attribute__((address_space(1))) int*)(p))
#define LASI(p) ((__attribute__((address_space(3))) int*)(p))
#define GAS4(p) ((__attribute__((address_space(1))) v4i*)(p))
#define LAS4(p) ((__attribute__((address_space(3))) v4i*)(p))
#define ASYNC_B32(g, l)  __builtin_amdgcn_global_load_async_to_lds_b32(GASI(g), LASI(l), 0, 0)
#define ASYNC_B128(g, l) __builtin_amdgcn_global_load_async_to_lds_b128(GAS4(g), LAS4(l), 0, 0)
#define WAIT_ASYNC() asm volatile("s_wait_asynccnt 0x0" ::: "memory")

constexpr int Bn = 8, N1 = 1024, N2 = 256, Dd = 16, Ee = 16;
constexpr int ROW = N2 * Dd;  // 4096 floats per (b,i) row

// LDS strides (floats)
constexpr int ASTR = 20;   // A-tile row stride: 8B aligned, 4-bank windows disjoint
constexpr int BSTR = 288;  // B-tile K-pair row stride: == 32 banks -> full 64-bank use

// ===========================================================================
// Kernel 1:  S0_b (1024 x 4096) = adj0^T (1024x1024) @ X_b (1024 x 4096)
// WG = 256 threads (8 waves), tile 128(M) x 128(N), K-tile 16, double-buffered
// async staging. Wave owns 2(M) x 4(N) 16x16 accumulators.
// ===========================================================================
__global__ __launch_bounds__(256) void tgcn_s0_gemm(const float* __restrict__ X,
                                                    const float* __restrict__ adj0,
                                                    float* __restrict__ S0) {
  __shared__ __align__(16) float As[2 * 128 * ASTR];  // adj0^T tile [m][k]
  __shared__ __align__(16) float Bs[2 * 8 * BSTR];    // X tile, K-pair interleaved

  const int tid   = threadIdx.x;
  const int lane  = tid & 31;
  const int wave  = tid >> 5;
  const int row_a = lane & 15;   // M (A) / N (B) lane coordinate
  const int half  = lane >> 4;   // K-pair select
  const int m_off = (wave & 3) * 32;
  const int n_off = (wave >> 2) * 64;
  const int m0    = blockIdx.y * 128;
  const int n0    = blockIdx.x * 128;
  const long boff = (long)blockIdx.z * (long)(N1 * ROW);

  const int cm = tid & 127;   // staging column (coalesced)
  const int ck = tid >> 7;    // staging k base (0..1)

  // issue async staging of one K-tile (16 rows of A^T and B) into buffer `bi`
  auto stage = [&](int kt, int bi) {
    float* Ab = As + bi * (128 * ASTR);
    float* Bb = Bs + bi * (8 * BSTR);
#pragma unroll
    for (int rep = 0; rep < 8; ++rep) {
      const int k = ck + rep * 2;
      // adj0 row (i = K) -> transposed into As[m][k]
      ASYNC_B32(adj0 + (long)(kt * 16 + k) * N1 + m0 + cm, Ab + cm * ASTR + k);
      // X row -> K-pair interleaved Bs[{k>>1}][n][k&1]
      ASYNC_B32(X + boff + (long)(kt * 16 + k) * ROW + n0 + cm,
                Bb + (k >> 1) * BSTR + cm * 2 + (k & 1));
    }
  };

  v8f acc[2][4] = {};

  stage(0, 0);
  for (int kt = 0; kt < N1 / 16; ++kt) {
    const int bi = kt & 1;
    WAIT_ASYNC();        // own asyncs done
    __syncthreads();     // everyone's asyncs done; prev buffer reads all retired
    if (kt + 1 < N1 / 16) stage(kt + 1, bi ^ 1);

    const float* Ab = As + bi * (128 * ASTR);
    const float* Bb = Bs + bi * (8 * BSTR);
#pragma unroll
    for (int kk = 0; kk < 16; kk += 4) {
      const int kb = kk + 2 * half;        // even
      const int p  = kb >> 1;
      v2f a0 = *(const v2f*)&Ab[(m_off + row_a) * ASTR + kb];
      v2f a1 = *(const v2f*)&Ab[(m_off + 16 + row_a) * ASTR + kb];
#pragma unroll
      for (int tn = 0; tn < 4; ++tn) {
        v2f bf = *(const v2f*)&Bb[p * BSTR + (n_off + tn * 16 + row_a) * 2];
        acc[0][tn] = WMMA_F32(a0, bf, acc[0][tn]);
        acc[1][tn] = WMMA_F32(a1, bf, acc[1][tn]);
      }
    }
  }

  // C/D layout: vgpr r -> row r + 8*half, col = row_a
#pragma unroll
  for (int tm = 0; tm < 2; ++tm)
#pragma unroll
    for (int tn = 0; tn < 4; ++tn)
#pragma unroll
      for (int r = 0; r < 8; ++r) {
        const int row = m0 + m_off + tm * 16 + r + 8 * half;
        const int col = n0 + n_off + tn * 16 + row_a;
        S0[boff + (long)row * ROW + col] = acc[tm][tn][r];
      }
}

// ===========================================================================
// Kernel 2: per (b,i):  P = X W + S0 Ws0 (regs),  Q = X Ws1 + S0 Ws01 (LDS),
//           R = adj1^T @ Q,  out = ReLU(P + R + bias_sum).
// WG = 256 threads (8 waves); wave w owns row-tiles {2w, 2w+1} (j and l).
// Async b128 staging for X/S0; async transpose + double buffer for adj1.
// ===========================================================================
__global__ __launch_bounds__(256) void tgcn_fuse(
    const float* __restrict__ X, const float* __restrict__ S0,
    const float* __restrict__ adj1,
    const float* __restrict__ W,   const float* __restrict__ bb,
    const float* __restrict__ Ws0, const float* __restrict__ bs0,
    const float* __restrict__ Ws1, const float* __restrict__ bs1,
    const float* __restrict__ Ws01, const float* __restrict__ bs01,
    float* __restrict__ out) {
  __shared__ __align__(16) float Xs[256 * ASTR];      // X_bi   [j][d]
  __shared__ __align__(16) float Ss[256 * ASTR];      // S0_bi  [j][d]
  __shared__ __align__(16) float Qf[128 * 32];        // Q, j-pair interleaved
  __shared__ __align__(16) float Adf[2 * 256 * ASTR]; // adj1^T slab [l][jj] x2
  __shared__ __align__(16) float Wf[4 * 256];         // weights, d-pair interleaved

  const int tid   = threadIdx.x;
  const int lane  = tid & 31;
  const int wave  = tid >> 5;
  const int row_a = lane & 15;
  const int half  = lane >> 4;
  const int i = blockIdx.x, b = blockIdx.y;
  const long base = ((long)b * N1 + i) * ROW;

  auto stageAdj = [&](int jt, int bi) {
    float* A = Adf + bi * (256 * ASTR);
#pragma unroll
    for (int jj = 0; jj < 16; ++jj)  // coalesced global, transposed LDS
      ASYNC_B32(adj1 + (jt * 16 + jj) * N2 + tid, A + tid * ASTR + jj);
  };

  // ---- stage X, S0 (async b128), weights (interleaved), adj tile 0 ---------
#pragma unroll
  for (int rep = 0; rep < 4; ++rep) {
    const int idx4 = rep * 256 + tid;
    const int j = idx4 >> 2, d4 = idx4 & 3;
    ASYNC_B128(X + base + idx4 * 4, Xs + j * ASTR + d4 * 4);
    ASYNC_B128(S0 + base + idx4 * 4, Ss + j * ASTR + d4 * 4);
  }
  {
    const int dd = tid >> 4, e = tid & 15;
    const int wofs = (dd >> 1) * 32 + e * 2 + (dd & 1);  // d-pair interleave
    Wf[0 * 256 + wofs] = W[tid];
    Wf[1 * 256 + wofs] = Ws0[tid];
    Wf[2 * 256 + wofs] = Ws1[tid];
    Wf[3 * 256 + wofs] = Ws01[tid];
  }
  stageAdj(0, 0);
  WAIT_ASYNC();
  __syncthreads();

  // ---- phase A: projections (K = D = 16) -----------------------------------
  v8f accP[2] = {};
  const int t0 = wave * 2;
#pragma unroll
  for (int t = 0; t < 2; ++t) {
    const int jbase = (t0 + t) * 16;
    v8f q = {};
#pragma unroll
    for (int kk = 0; kk < 16; kk += 4) {
      const int kb = kk + 2 * half;
      const int p  = kb >> 1;
      v2f ax = *(const v2f*)&Xs[(jbase + row_a) * ASTR + kb];
      v2f as = *(const v2f*)&Ss[(jbase + row_a) * ASTR + kb];
      v2f bw   = *(const v2f*)&Wf[0 * 256 + p * 32 + row_a * 2];
      v2f bw0  = *(const v2f*)&Wf[1 * 256 + p * 32 + row_a * 2];
      v2f bw1  = *(const v2f*)&Wf[2 * 256 + p * 32 + row_a * 2];
      v2f bw01 = *(const v2f*)&Wf[3 * 256 + p * 32 + row_a * 2];
      accP[t] = WMMA_F32(ax, bw,  accP[t]);
      accP[t] = WMMA_F32(as, bw0, accP[t]);
      q = WMMA_F32(ax, bw1,  q);
      q = WMMA_F32(as, bw01, q);
    }
#pragma unroll
    for (int r = 0; r < 8; ++r) {  // j-pair interleaved store of Q
      const int j = jbase + r + 8 * half;
      Qf[(j >> 1) * 32 + row_a * 2 + (r & 1)] = q[r];
    }
  }

  // ---- phase B: R = adj1^T @ Q, K = 256, double-buffered adj staging -------
  v8f accR[2] = {};
  const int lb0 = t0 * 16;
  for (int jt = 0; jt < 16; ++jt) {
    const int bi = jt & 1;
    WAIT_ASYNC();
    __syncthreads();  // adj tile ready everywhere; Qs visible (jt==0)
    if (jt + 1 < 16) stageAdj(jt + 1, bi ^ 1);

    const float* Ab = Adf + bi * (256 * ASTR);
    const int j0 = jt * 16;
#pragma unroll
    for (int t = 0; t < 2; ++t) {
      const int lb = lb0 + t * 16;
#pragma unroll
      for (int kk = 0; kk < 16; kk += 4) {
        const int kb = kk + 2 * half;
        v2f a  = *(const v2f*)&Ab[(lb + row_a) * ASTR + kb];
        v2f bf = *(const v2f*)&Qf[((j0 + kb) >> 1) * 32 + row_a * 2];
        accR[t] = WMMA_F32(a, bf, accR[t]);
      }
    }
  }

  // ---- epilogue: bias + ReLU + store ---------------------------------------
  const float bias = bb[row_a] + bs0[row_a] + bs1[row_a] + bs01[row_a];
#pragma unroll
  for (int t = 0; t < 2; ++t) {
    const int lb = lb0 + t * 16;
#pragma unroll
    for (int r = 0; r < 8; ++r) {
      float v = accP[t][r] + accR[t][r] + bias;
      v = v > 0.0f ? v : 0.0f;
      out[base + (long)(lb + r + 8 * half) * Ee + row_a] = v;
    }
  }
}

// ===========================================================================
extern "C" void kernel_launch(void* const* d_in, const int* in_sizes, int n_in,
                              void* d_out, int out_size, void* d_ws, size_t ws_size,
                              hipStream_t stream) {
  const float* X    = (const float*)d_in[0];
  const float* adj0 = (const float*)d_in[1];
  const float* adj1 = (const float*)d_in[2];
  const float* W    = (const float*)d_in[3];
  const float* bb   = (const float*)d_in[4];
  const float* Ws0  = (const float*)d_in[5];
  const float* bs0  = (const float*)d_in[6];
  const float* Ws1  = (const float*)d_in[7];
  const float* bs1  = (const float*)d_in[8];
  const float* Ws01 = (const float*)d_in[9];
  const float* bs01 = (const float*)d_in[10];
  float* out = (float*)d_out;
  float* S0  = (float*)d_ws;  // 8*1024*256*16 floats = 128 MiB scratch

  // Kernel 1: per-b GEMM S0 = adj0^T @ X_b   (grid: N/128 x M/128 x B)
  tgcn_s0_gemm<<<dim3(32, 8, 8), 256, 0, stream>>>(X, adj0, S0);

  // Kernel 2: fused projection + mode-1 aggregation + ReLU (one WG per (b,i))
  tgcn_fuse<<<dim3(1024, 8), 256, 0, stream>>>(X, S0, adj1, W, bb, Ws0, bs0,
                                               Ws1, bs1, Ws01, bs01, out);
}